// DHCN_77695958385170
// MI455X (gfx1250) — compile-verified
//
#include <hip/hip_runtime.h>
#include <hip/hip_bf16.h>

#define N_NODE  200000
#define HIDDEN  128
#define BATCH   1024
#define SEQ     50
#define NNZ     3200000
#define N_EMB   (N_NODE * HIDDEN)   // 25,600,000

typedef float v2f __attribute__((ext_vector_type(2)));
typedef float v8f __attribute__((ext_vector_type(8)));

// ---------------------------------------------------------------------------
// Elementwise helpers
// ---------------------------------------------------------------------------
__global__ void vec_copy(float* __restrict__ y, const float* __restrict__ x, int n) {
    int i = blockIdx.x * blockDim.x + threadIdx.x;
    if (i < n) y[i] = x[i];
}
__global__ void vec_zero(float* __restrict__ y, int n) {
    int i = blockIdx.x * blockDim.x + threadIdx.x;
    if (i < n) y[i] = 0.0f;
}
__global__ void vec_add(float* __restrict__ y, const float* __restrict__ x, int n) {
    int i = blockIdx.x * blockDim.x + threadIdx.x;
    if (i < n) y[i] += x[i];
}
__global__ void vec_add_scale(float* __restrict__ y, const float* __restrict__ x, float s, int n) {
    int i = blockIdx.x * blockDim.x + threadIdx.x;
    if (i < n) y[i] = (y[i] + x[i]) * s;
}
__global__ void vec_scale_copy(float* __restrict__ y, const float* __restrict__ x, float s, int n) {
    int i = blockIdx.x * blockDim.x + threadIdx.x;
    if (i < n) y[i] = x[i] * s;
}

// ---------------------------------------------------------------------------
// Hypergraph SpMV scatter: next[rows[e], h] += vals[e] * cur[cols[e], h]
// One thread per (edge, h); a wave's 32 lanes share one edge -> scalar loads of
// rows/cols/vals broadcast, fully coalesced 128B gather & atomic scatter.
// ---------------------------------------------------------------------------
__global__ void spmv_scatter(const int* __restrict__ rows, const int* __restrict__ cols,
                             const float* __restrict__ vals, const float* __restrict__ cur,
                             float* __restrict__ next) {
    int tid = blockIdx.x * blockDim.x + threadIdx.x;     // < NNZ*128 = 409.6M
    int e = tid >> 7;
    int h = tid & 127;
    if (e >= NNZ) return;
    int r = rows[e];
    int c = cols[e];
    float v = vals[e];
    atomicAdd(&next[(size_t)r * HIDDEN + h], v * cur[(size_t)c * HIDDEN + h]);
}

// ---------------------------------------------------------------------------
// Dense fp32 GEMM using V_WMMA_F32_16X16X4_F32, one wave per 16x16 C tile.
// A-matrix (16x4): lanes 0-15 -> M=lane, VGPR0=K0,VGPR1=K1; lanes 16-31 -> K2,K3
// B-matrix (4x16): lanes 0-15 -> N=lane, rows K0/K1; lanes 16-31 -> rows K2/K3
// C/D (16x16):     VGPR j: lanes 0-15 -> M=j, N=lane; lanes 16-31 -> M=j+8
// M,N multiples of 16; K multiple of 4.
// ---------------------------------------------------------------------------
__global__ void wmma_gemm_f32(const float* __restrict__ A, const float* __restrict__ B,
                              float* __restrict__ C, int M, int N, int K,
                              int lda, int ldb, int ldc) {
    int gtid   = blockIdx.x * blockDim.x + threadIdx.x;
    int wave   = gtid >> 5;                 // wave-uniform
    int lane   = threadIdx.x & 31;
    int tilesN = N >> 4;
    int tilesM = M >> 4;
    if (wave >= tilesM * tilesN) return;    // uniform per wave: EXEC all-1 inside
    int tm   = (wave / tilesN) << 4;
    int tn   = (wave % tilesN) << 4;
    int half = lane >> 4;                   // 0 or 1
    int l15  = lane & 15;

    v8f acc = {0.f, 0.f, 0.f, 0.f, 0.f, 0.f, 0.f, 0.f};
    const float* arow = A + (size_t)(tm + l15) * lda;
    for (int k = 0; k < K; k += 4) {
        int ka = k + 2 * half;
        v2f a, b;
        a.x = arow[ka];
        a.y = arow[ka + 1];
        b.x = B[(size_t)ka       * ldb + tn + l15];
        b.y = B[(size_t)(ka + 1) * ldb + tn + l15];
        acc = __builtin_amdgcn_wmma_f32_16x16x4_f32(
                  false, a, false, b, (short)0, acc, false, false);
    }
    int n     = tn + l15;
    int mbase = tm + (half << 3);
#pragma unroll
    for (int j = 0; j < 8; ++j)
        C[(size_t)(mbase + j) * ldc + n] = acc[j];
}

// ---------------------------------------------------------------------------
// sess[b,h] = (1/len[b]) * sum_l padded[info[b,l]]  where padded[0]=0,
// padded[i]=emb[i-1].  One block per b, one thread per h.
// ---------------------------------------------------------------------------
__global__ void session_mean(const float* __restrict__ emb, const long long* __restrict__ info,
                             const int* __restrict__ slen, float* __restrict__ out) {
    int b = blockIdx.x;
    int h = threadIdx.x;
    float s = 0.0f;
    for (int l = 0; l < SEQ; ++l) {
        long long idx = info[(size_t)b * SEQ + l];
        if (idx != 0) s += emb[(size_t)(idx - 1) * HIDDEN + h];
    }
    out[(size_t)b * HIDDEN + h] = s / (float)slen[b];
}

// new_seq[b,l,h] = tanh( [pos[l], seq_h[b,l]] @ w1_W[:,h] + w1_b[h] )
__global__ void w1_kernel(const float* __restrict__ hg_item, const long long* __restrict__ rinfo,
                          const float* __restrict__ pos_tab, const float* __restrict__ W,
                          const float* __restrict__ bvec, float* __restrict__ new_seq) {
    int bl = blockIdx.x;            // b*SEQ + l
    int l  = bl % SEQ;
    int h  = threadIdx.x;           // 128
    __shared__ float x[2 * HIDDEN];
    long long idx = rinfo[bl];
    x[h]          = pos_tab[(size_t)l * HIDDEN + h];
    x[HIDDEN + h] = (idx != 0) ? hg_item[(size_t)(idx - 1) * HIDDEN + h] : 0.0f;
    __syncthreads();
    float s = bvec[h];
    for (int k = 0; k < 2 * HIDDEN; ++k)
        s += x[k] * W[(size_t)k * HIDDEN + h];
    new_seq[(size_t)bl * HIDDEN + h] = tanhf(s);
}

// m2[b,h] = sess_mean[b] @ w2_W[:,h] + w2_b[h]
__global__ void w2_kernel(const float* __restrict__ sess_mean, const float* __restrict__ W,
                          const float* __restrict__ bvec, float* __restrict__ out) {
    int b = blockIdx.x;
    int h = threadIdx.x;
    __shared__ float x[HIDDEN];
    x[h] = sess_mean[(size_t)b * HIDDEN + h];
    __syncthreads();
    float s = bvec[h];
    for (int k = 0; k < HIDDEN; ++k)
        s += x[k] * W[(size_t)k * HIDDEN + h];
    out[(size_t)b * HIDDEN + h] = s;
}

// alpha[b,l] = mask * sum_h sigmoid(m2[b,h] + new_seq[b,l]@w3_W[:,h] + w3_b[h]) * f_T[h]
__global__ void alpha_kernel(const float* __restrict__ new_seq, const float* __restrict__ m2,
                             const float* __restrict__ W, const float* __restrict__ bvec,
                             const float* __restrict__ fT, const unsigned char* __restrict__ masks,
                             float* __restrict__ alpha) {
    int bl = blockIdx.x;
    int b  = bl / SEQ;
    int h  = threadIdx.x;
    __shared__ float x[HIDDEN];
    __shared__ float red[HIDDEN];
    x[h] = new_seq[(size_t)bl * HIDDEN + h];
    __syncthreads();
    float s = bvec[h] + m2[(size_t)b * HIDDEN + h];
    for (int k = 0; k < HIDDEN; ++k)
        s += x[k] * W[(size_t)k * HIDDEN + h];
    float g = 1.0f / (1.0f + expf(-s));
    red[h] = g * fT[h];
    __syncthreads();
    for (int off = 64; off > 0; off >>= 1) {
        if (h < off) red[h] += red[h + off];
        __syncthreads();
    }
    if (h == 0) alpha[bl] = red[0] * (masks[bl] ? 1.0f : 0.0f);
}

// hg_sess[b,h] = sum_l alpha[b,l] * new_seq[b,l,h]
__global__ void theta_kernel(const float* __restrict__ new_seq, const float* __restrict__ alpha,
                             float* __restrict__ hg_sess) {
    int b = blockIdx.x;
    int h = threadIdx.x;
    float s = 0.0f;
    for (int l = 0; l < SEQ; ++l)
        s += alpha[b * SEQ + l] * new_seq[((size_t)b * SEQ + l) * HIDDEN + h];
    hg_sess[(size_t)b * HIDDEN + h] = s;
}

// per-batch SSL contribution (deterministic two-stage reduction)
__global__ void ssl_kernel(const float* __restrict__ hg_sess, const float* __restrict__ line_sess,
                           const int* __restrict__ perm_row, const int* __restrict__ perm_col,
                           float* __restrict__ part) {
    int b = blockIdx.x;
    int h = threadIdx.x;
    __shared__ float r1[HIDDEN];
    __shared__ float r2[HIDDEN];
    float ls = line_sess[(size_t)b * HIDDEN + h];
    r1[h] = hg_sess[(size_t)b * HIDDEN + h] * ls;
    r2[h] = hg_sess[(size_t)perm_row[b] * HIDDEN + perm_col[h]] * ls;
    __syncthreads();
    for (int off = 64; off > 0; off >>= 1) {
        if (h < off) { r1[h] += r1[h + off]; r2[h] += r2[h + off]; }
        __syncthreads();
    }
    if (h == 0) {
        const float eps = 1e-8f;
        float sp = 1.0f / (1.0f + expf(-r1[0]));
        float sn = 1.0f / (1.0f + expf(-r2[0]));
        part[b] = -logf(eps + sp) - logf(eps + (1.0f - sn));
    }
}
__global__ void ssl_reduce(const float* __restrict__ part, float* __restrict__ out) {
    __shared__ float red[256];
    int t = threadIdx.x;
    float s = 0.0f;
    for (int i = t; i < BATCH; i += 256) s += part[i];
    red[t] = s;
    __syncthreads();
    for (int off = 128; off > 0; off >>= 1) {
        if (t < off) red[t] += red[t + off];
        __syncthreads();
    }
    if (t == 0) *out = 0.01f * red[0];   // BETA
}

// ---------------------------------------------------------------------------
extern "C" void kernel_launch(void* const* d_in, const int* in_sizes, int n_in,
                              void* d_out, int out_size, void* d_ws, size_t ws_size,
                              hipStream_t stream) {
    const float*     items    = (const float*)d_in[0];
    const float*     pos_tab  = (const float*)d_in[1];
    const float*     w1W      = (const float*)d_in[2];
    const float*     w1b      = (const float*)d_in[3];
    const float*     w2W      = (const float*)d_in[4];
    const float*     w2b      = (const float*)d_in[5];
    const float*     w3W      = (const float*)d_in[6];
    const float*     w3b      = (const float*)d_in[7];
    const float*     fT       = (const float*)d_in[8];
    const float*     hg_vals  = (const float*)d_in[9];
    const float*     line_adj = (const float*)d_in[10];
    const float*     degree   = (const float*)d_in[11];
    const int*       hg_rows  = (const int*)d_in[12];
    const int*       hg_cols  = (const int*)d_in[13];
    const long long* sess_inf = (const long long*)d_in[14];
    const long long* rev_inf  = (const long long*)d_in[15];
    const int*       slen     = (const int*)d_in[16];
    const unsigned char* masks = (const unsigned char*)d_in[17];
    const int*       perm_row = (const int*)d_in[18];
    const int*       perm_col = (const int*)d_in[19];

    float* out     = (float*)d_out;
    float* hg_sess = out;                          // [1024,128]
    float* loss    = out + BATCH * HIDDEN;         // [1]
    float* hg_item = out + BATCH * HIDDEN + 1;     // [200000,128] (4B aligned only)

    float* ws   = (float*)d_ws;
    float* bufA = ws;                              // 25.6M floats
    float* bufB = ws + (size_t)N_EMB;              // 25.6M floats
    // Aliases inside bufA's region — valid only after hypergraph conv is done:
    float* DA        = ws;                                     // 1,048,576
    float* sess_mean = ws + 1048576;                           // 131,072
    float* m2        = sess_mean + BATCH * HIDDEN;             // 131,072
    float* new_seq   = m2 + BATCH * HIDDEN;                    // 6,553,600
    float* alpha     = new_seq + (size_t)BATCH * SEQ * HIDDEN; // 51,200
    float* line_cur  = alpha + BATCH * SEQ;
    float* line_next = line_cur + BATCH * HIDDEN;
    float* line_acc  = line_next + BATCH * HIDDEN;
    float* line_sess = line_acc + BATCH * HIDDEN;
    float* loss_part = line_sess + BATCH * HIDDEN;             // 1024

    const int T = 256;
    const int embBlocks  = (N_EMB + T - 1) / T;
    const int spmvBlocks = (int)(((long long)NNZ * HIDDEN) / T);

    // ---- 1. Hypergraph conv: acc = (x + Ax + A^2x + A^3x)/4  -> hg_item ----
    vec_copy<<<embBlocks, T, 0, stream>>>(hg_item, items, N_EMB);

    vec_zero<<<embBlocks, T, 0, stream>>>(bufA, N_EMB);
    spmv_scatter<<<spmvBlocks, T, 0, stream>>>(hg_rows, hg_cols, hg_vals, items, bufA);
    vec_add<<<embBlocks, T, 0, stream>>>(hg_item, bufA, N_EMB);

    vec_zero<<<embBlocks, T, 0, stream>>>(bufB, N_EMB);
    spmv_scatter<<<spmvBlocks, T, 0, stream>>>(hg_rows, hg_cols, hg_vals, bufA, bufB);
    vec_add<<<embBlocks, T, 0, stream>>>(hg_item, bufB, N_EMB);

    vec_zero<<<embBlocks, T, 0, stream>>>(bufA, N_EMB);
    spmv_scatter<<<spmvBlocks, T, 0, stream>>>(hg_rows, hg_cols, hg_vals, bufB, bufA);
    vec_add_scale<<<embBlocks, T, 0, stream>>>(hg_item, bufA, 0.25f, N_EMB);
    // bufA/bufB free from here on (aliased scratch becomes usable).

    // ---- 2. Attention branch -> hg_sess ----
    session_mean<<<BATCH, HIDDEN, 0, stream>>>(hg_item, rev_inf, slen, sess_mean);
    w2_kernel<<<BATCH, HIDDEN, 0, stream>>>(sess_mean, w2W, w2b, m2);
    w1_kernel<<<BATCH * SEQ, HIDDEN, 0, stream>>>(hg_item, rev_inf, pos_tab, w1W, w1b, new_seq);
    alpha_kernel<<<BATCH * SEQ, HIDDEN, 0, stream>>>(new_seq, m2, w3W, w3b, fT, masks, alpha);
    theta_kernel<<<BATCH, HIDDEN, 0, stream>>>(new_seq, alpha, hg_sess);

    // ---- 3. Line-graph conv (WMMA fp32 GEMMs) -> line_sess ----
    session_mean<<<BATCH, HIDDEN, 0, stream>>>(items, sess_inf, slen, line_cur);
    vec_copy<<<(BATCH * HIDDEN + T - 1) / T, T, 0, stream>>>(line_acc, line_cur, BATCH * HIDDEN);

    {   // DA = degree @ line_adj  (1024x1024x1024)
        int tiles = (BATCH / 16) * (BATCH / 16);
        wmma_gemm_f32<<<(tiles * 32 + T - 1) / T, T, 0, stream>>>(
            degree, line_adj, DA, BATCH, BATCH, BATCH, BATCH, BATCH, BATCH);
    }
    float* cur = line_cur;
    float* nxt = line_next;
    int ptiles = (BATCH / 16) * (HIDDEN / 16);
    for (int step = 0; step < 3; ++step) {
        wmma_gemm_f32<<<(ptiles * 32 + T - 1) / T, T, 0, stream>>>(
            DA, cur, nxt, BATCH, HIDDEN, BATCH, BATCH, HIDDEN, HIDDEN);
        vec_add<<<(BATCH * HIDDEN + T - 1) / T, T, 0, stream>>>(line_acc, nxt, BATCH * HIDDEN);
        float* t = cur; cur = nxt; nxt = t;
    }
    vec_scale_copy<<<(BATCH * HIDDEN + T - 1) / T, T, 0, stream>>>(
        line_sess, line_acc, 0.25f, BATCH * HIDDEN);

    // ---- 4. SSL loss ----
    ssl_kernel<<<BATCH, HIDDEN, 0, stream>>>(hg_sess, line_sess, perm_row, perm_col, loss_part);
    ssl_reduce<<<1, 256, 0, stream>>>(loss_part, loss);
}